// EdgeBlock_82394652606663
// MI455X (gfx1250) — compile-verified
//
#include <hip/hip_runtime.h>

typedef __attribute__((ext_vector_type(2))) float v2f;
typedef __attribute__((ext_vector_type(8))) float v8f;

#define D_NODE   64
#define D_EDGE   64
#define D_GLOBAL 32
#define D_IN     224   // K = 64 + 64 + 64 + 32
#define D_OUT    64
#define W_STRIDE 228   // padded K-stride for transposed W in LDS (conflict-free)
#define WAVES_PER_BLOCK 8
#define BLOCK_THREADS (WAVES_PER_BLOCK * 32)

__global__ __launch_bounds__(BLOCK_THREADS)
void edge_block_wmma_f32(const float* __restrict__ edges,
                         const float* __restrict__ nodes,
                         const float* __restrict__ globals_vec,
                         const int*   __restrict__ receivers,
                         const int*   __restrict__ senders,
                         const float* __restrict__ W,
                         const float* __restrict__ bias,
                         float* __restrict__ out,
                         int n_edges, int n_tiles)
{
    // Transposed W in LDS: sW[n * W_STRIDE + k] = W[k*64 + n]
    __shared__ float sW[D_OUT * W_STRIDE];
    __shared__ float sB[D_OUT];

    for (int idx = threadIdx.x; idx < D_IN * D_OUT; idx += BLOCK_THREADS) {
        int k = idx >> 6;     // 0..223
        int n = idx & 63;     // 0..63
        sW[n * W_STRIDE + k] = W[idx];
    }
    if (threadIdx.x < D_OUT) sB[threadIdx.x] = bias[threadIdx.x];
    __syncthreads();

    const int lane = threadIdx.x & 31;
    const int m    = lane & 15;          // A row within tile / B,C column within n-tile
    const int kh   = (lane >> 4) * 2;    // lane's K sub-offset within a k-step: 0 or 2

    const int waveGlobal = blockIdx.x * WAVES_PER_BLOCK + (threadIdx.x >> 5);
    const int waveCount  = gridDim.x * WAVES_PER_BLOCK;

    // Bias values for this lane's output column in each of the 4 n-tiles.
    const float bv0 = sB[m + 0];
    const float bv1 = sB[m + 16];
    const float bv2 = sB[m + 32];
    const float bv3 = sB[m + 48];

    // Globals A fragments are identical for every tile: hoist (8 pairs = 16 VGPRs).
    v2f aG[8];
#pragma unroll
    for (int kk = 0; kk < 8; ++kk)
        aG[kk] = *(const v2f*)(globals_vec + kk * 4 + kh);

    for (int tile = waveGlobal; tile < n_tiles; tile += waveCount) {
        // Opaque zero: forces B-fragment LDS loads to be re-issued per tile,
        // preventing LICM from hoisting 448 VGPRs of B across the tile loop
        // (which spilled to scratch).
        unsigned swo = 0;
        asm volatile("" : "+v"(swo));
        const float* sWt = sW + swo;

        int e  = tile * 16 + m;
        int ec = e < n_edges ? e : (n_edges - 1);
        int r = receivers[ec];
        int s = senders[ec];

        const float* edge_row = edges + (size_t)ec * D_EDGE;
        const float* recv_row = nodes + (size_t)r * D_NODE;
        const float* send_row = nodes + (size_t)s * D_NODE;

        v8f c0 = {}, c1 = {}, c2 = {}, c3 = {};

        // One k-step: 4 B fragments from LDS + 4 WMMAs against a shared A pair.
        auto bstep = [&](v2f a, int kl) {
            v2f b0 = *(const v2f*)(sWt + (m +  0) * W_STRIDE + kl);
            v2f b1 = *(const v2f*)(sWt + (m + 16) * W_STRIDE + kl);
            v2f b2 = *(const v2f*)(sWt + (m + 32) * W_STRIDE + kl);
            v2f b3 = *(const v2f*)(sWt + (m + 48) * W_STRIDE + kl);
            c0 = __builtin_amdgcn_wmma_f32_16x16x4_f32(false, a, false, b0, (short)0, c0, false, false);
            c1 = __builtin_amdgcn_wmma_f32_16x16x4_f32(false, a, false, b1, (short)0, c1, false, false);
            c2 = __builtin_amdgcn_wmma_f32_16x16x4_f32(false, a, false, b2, (short)0, c2, false, false);
            c3 = __builtin_amdgcn_wmma_f32_16x16x4_f32(false, a, false, b3, (short)0, c3, false, false);
        };

        v2f aBuf[16];

        // ---- segment 0: edges (K 0..63), streaming -> non-temporal ----
#pragma unroll
        for (int kk = 0; kk < 16; ++kk)
            aBuf[kk] = __builtin_nontemporal_load((const v2f*)(edge_row + kk * 4 + kh));
#pragma unroll
        for (int kk = 0; kk < 16; ++kk)
            bstep(aBuf[kk], kk * 4 + kh);

        // ---- segment 1: receiver nodes (K 64..127), gather -> keep in L2 ----
#pragma unroll
        for (int kk = 0; kk < 16; ++kk)
            aBuf[kk] = *(const v2f*)(recv_row + kk * 4 + kh);
#pragma unroll
        for (int kk = 0; kk < 16; ++kk)
            bstep(aBuf[kk], 64 + kk * 4 + kh);

        // ---- segment 2: sender nodes (K 128..191) ----
#pragma unroll
        for (int kk = 0; kk < 16; ++kk)
            aBuf[kk] = *(const v2f*)(send_row + kk * 4 + kh);
#pragma unroll
        for (int kk = 0; kk < 16; ++kk)
            bstep(aBuf[kk], 128 + kk * 4 + kh);

        // ---- segment 3: globals (K 192..223), preloaded ----
#pragma unroll
        for (int kk = 0; kk < 8; ++kk)
            bstep(aG[kk], 192 + kk * 4 + kh);

        // C/D layout: VGPR j -> (lanes 0-15: row j, N=lane), (lanes 16-31: row j+8, N=lane-16)
        const int mb = (lane >> 4) * 8;
#pragma unroll
        for (int j = 0; j < 8; ++j) {
            int row = tile * 16 + mb + j;
            if (row < n_edges) {
                float* orow = out + (size_t)row * D_OUT;
                __builtin_nontemporal_store(c0[j] + bv0, orow + m +  0);
                __builtin_nontemporal_store(c1[j] + bv1, orow + m + 16);
                __builtin_nontemporal_store(c2[j] + bv2, orow + m + 32);
                __builtin_nontemporal_store(c3[j] + bv3, orow + m + 48);
            }
        }
    }
}

extern "C" void kernel_launch(void* const* d_in, const int* in_sizes, int n_in,
                              void* d_out, int out_size, void* d_ws, size_t ws_size,
                              hipStream_t stream) {
    const float* edges       = (const float*)d_in[0];
    const float* nodes       = (const float*)d_in[1];
    const float* globals_vec = (const float*)d_in[2];
    const int*   receivers   = (const int*)d_in[3];
    const int*   senders     = (const int*)d_in[4];
    const float* W           = (const float*)d_in[5];
    const float* b           = (const float*)d_in[6];
    float*       out         = (float*)d_out;

    int n_edges = in_sizes[0] / D_EDGE;
    int n_tiles = (n_edges + 15) / 16;

    int blocks = (n_tiles + WAVES_PER_BLOCK - 1) / WAVES_PER_BLOCK;
    if (blocks > 2048) blocks = 2048;

    edge_block_wmma_f32<<<blocks, BLOCK_THREADS, 0, stream>>>(
        edges, nodes, globals_vec, receivers, senders, W, b, out, n_edges, n_tiles);
}